// LSTMLinear_16140487098917
// MI455X (gfx1250) — compile-verified
//
#include <hip/hip_runtime.h>

typedef float v2f __attribute__((ext_vector_type(2)));
typedef float v8f __attribute__((ext_vector_type(8)));

#define LOG2E 1.4426950408889634f

__device__ __forceinline__ float dswz16(float v) {
  // SWAPX16: group-of-32, xor_mask=0x10, and_mask=0x1f -> offset 0x401f
  return __int_as_float(__builtin_amdgcn_ds_swizzle(__float_as_int(v), 0x401f));
}

__device__ __forceinline__ float fexp2(float x) {
#if __has_builtin(__builtin_amdgcn_exp2f)
  return __builtin_amdgcn_exp2f(x);
#else
  return exp2f(x);
#endif
}
__device__ __forceinline__ float frcp(float x) {
#if __has_builtin(__builtin_amdgcn_rcpf)
  return __builtin_amdgcn_rcpf(x);
#else
  return 1.0f / x;
#endif
}
__device__ __forceinline__ float flog2(float x) {
#if __has_builtin(__builtin_amdgcn_logf)
  return __builtin_amdgcn_logf(x);
#else
  return log2f(x);
#endif
}
__device__ __forceinline__ float fsig(float x) {
  return frcp(1.0f + fexp2(x * -LOG2E));
}
__device__ __forceinline__ float ftanh(float x) {
#if __has_builtin(__builtin_amdgcn_tanhf)
  return __builtin_amdgcn_tanhf(x);
#elif __has_builtin(__builtin_amdgcn_tanh_f32)
  return __builtin_amdgcn_tanh_f32(x);
#else
  // tanh(x) = 1 - 2/(exp(2x)+1)
  float e = fexp2(x * (2.0f * LOG2E));
  return __builtin_fmaf(-2.0f, frcp(e + 1.0f), 1.0f);
#endif
}

// One wave32 = 16 batch rows. D tile: M=gate(0..15 within gate-group tile),
// N=batch column. Lane L (<16) holds batch b0+L, hidden j=0..7 across the 8
// C/D VGPRs; lane L+16 holds j=8..15.
__global__ __launch_bounds__(256) void lstm_head_fused(
    const float* __restrict__ x,     // [B, T]
    const float* __restrict__ W_ih,  // [64]
    const float* __restrict__ W_hh,  // [64,16]
    const float* __restrict__ b_ih,  // [64]
    const float* __restrict__ b_hh,  // [64]
    const float* __restrict__ W1,    // [32,16]
    const float* __restrict__ b1,    // [32]
    const float* __restrict__ W2,    // [6,32]
    const float* __restrict__ b2,    // [6]
    float* __restrict__ out,         // [B,6] log-probs
    int T) {
  const int lane = threadIdx.x & 31;
  const int wv = threadIdx.x >> 5;
  const int row = lane & 15;
  const bool hi = lane >= 16;
  const int b = (blockIdx.x * 8 + wv) * 16 + row;

  // A = W_hh, 4 gate tiles x 4 K-chunks, kept in VGPRs.
  // A 16x4 f32 layout: vgpr0 = (K=4ck+0 | K=4ck+2), vgpr1 = (K=4ck+1 | K=4ck+3), M = lane%16.
  v2f A[4][4];
#pragma unroll
  for (int tt = 0; tt < 4; ++tt) {
#pragma unroll
    for (int ck = 0; ck < 4; ++ck) {
      const float* wr = W_hh + (16 * tt + row) * 16 + 4 * ck + (hi ? 2 : 0);
      v2f a;
      a.x = wr[0];
      a.y = wr[1];
      A[tt][ck] = a;
    }
  }

  // Per-lane seed constants: D-layout gate index g = 16*tt + k + (hi?8:0).
  float wih[4][8], bsum[4][8];
#pragma unroll
  for (int tt = 0; tt < 4; ++tt) {
#pragma unroll
    for (int k = 0; k < 8; ++k) {
      int g = 16 * tt + k + (hi ? 8 : 0);
      wih[tt][k] = W_ih[g];
      bsum[tt][k] = b_ih[g] + b_hh[g];
    }
  }

  float hj[8], cst[8];
#pragma unroll
  for (int k = 0; k < 8; ++k) {
    hj[k] = 0.0f;
    cst[k] = 0.0f;
  }

  const float* xr = x + (size_t)b * T;

  for (int t0 = 0; t0 < T; t0 += 4) {
    float4 xq = *(const float4*)(xr + t0);
    __builtin_prefetch(xr + t0 + 64, 0, 0);  // global_prefetch ahead in the row
#pragma unroll
    for (int u = 0; u < 4; ++u) {
      float xv = (u == 0) ? xq.x : (u == 1) ? xq.y : (u == 2) ? xq.z : xq.w;

      // C seed: xg = x*W_ih + (b_ih+b_hh), already in D layout.
      v8f acc[4];
#pragma unroll
      for (int tt = 0; tt < 4; ++tt) {
#pragma unroll
        for (int k = 0; k < 8; ++k)
          acc[tt][k] = __builtin_fmaf(xv, wih[tt][k], bsum[tt][k]);
      }

      // Build B = h^T in 4x16 f32 layout from lane-local h:
      // lane L needs K={4ck,4ck+1}, lane L+16 needs K={4ck+2,4ck+3}.
      float s0 = dswz16(hj[0]), s1 = dswz16(hj[1]), s2 = dswz16(hj[2]),
            s3 = dswz16(hj[3]), s4 = dswz16(hj[4]), s5 = dswz16(hj[5]),
            s6 = dswz16(hj[6]), s7 = dswz16(hj[7]);
      v2f Bv[4];
      Bv[0].x = hi ? s2 : hj[0];
      Bv[0].y = hi ? s3 : hj[1];
      Bv[1].x = hi ? s6 : hj[4];
      Bv[1].y = hi ? s7 : hj[5];
      Bv[2].x = hi ? hj[2] : s0;
      Bv[2].y = hi ? hj[3] : s1;
      Bv[3].x = hi ? hj[6] : s4;
      Bv[3].y = hi ? hj[7] : s5;

      // gates^T += W_hh(tile) x h^T : 4 chained K-chunks per gate tile.
#pragma unroll
      for (int tt = 0; tt < 4; ++tt) {
#pragma unroll
        for (int ck = 0; ck < 4; ++ck) {
          acc[tt] = __builtin_amdgcn_wmma_f32_16x16x4_f32(
              false, A[tt][ck], false, Bv[ck], (short)0, acc[tt], false, false);
        }
      }

      // Lane-local LSTM cell update (gate order i,f,g,o).
#pragma unroll
      for (int k = 0; k < 8; ++k) {
        float ig = fsig(acc[0][k]);
        float fg = fsig(acc[1][k]);
        float gg = ftanh(acc[2][k]);
        float og = fsig(acc[3][k]);
        float c = __builtin_fmaf(fg, cst[k], ig * gg);
        cst[k] = c;
        hj[k] = og * ftanh(c);
      }
    }
  }

  // ---- fused head: relu(h @ W1^T + b1) @ W2^T + b2, then log_softmax ----
  const int joff = hi ? 8 : 0;
  float r[32];
#pragma unroll
  for (int o = 0; o < 32; ++o) {
    float p = 0.0f;
#pragma unroll
    for (int k = 0; k < 8; ++k)
      p = __builtin_fmaf(hj[k], W1[o * 16 + joff + k], p);
    p += dswz16(p);  // combine j=0..7 (lane L) with j=8..15 (lane L+16)
    p += b1[o];
    r[o] = p > 0.0f ? p : 0.0f;
  }
  float z[6];
#pragma unroll
  for (int j = 0; j < 6; ++j) {
    float a = b2[j];
#pragma unroll
    for (int o = 0; o < 32; ++o) a = __builtin_fmaf(r[o], W2[j * 32 + o], a);
    z[j] = a;
  }
  float m = z[0];
#pragma unroll
  for (int j = 1; j < 6; ++j) m = fmaxf(m, z[j]);
  float s = 0.0f;
#pragma unroll
  for (int j = 0; j < 6; ++j) s += fexp2((z[j] - m) * LOG2E);
  float lse = m + flog2(s) * (1.0f / LOG2E);
  if (!hi) {
#pragma unroll
    for (int j = 0; j < 6; ++j) out[b * 6 + j] = z[j] - lse;
  }
}

extern "C" void kernel_launch(void* const* d_in, const int* in_sizes, int n_in,
                              void* d_out, int out_size, void* d_ws,
                              size_t ws_size, hipStream_t stream) {
  (void)n_in;
  (void)d_ws;
  (void)ws_size;
  const float* x = (const float*)d_in[0];
  const float* W_ih = (const float*)d_in[1];
  const float* W_hh = (const float*)d_in[2];
  const float* b_ih = (const float*)d_in[3];
  const float* b_hh = (const float*)d_in[4];
  const float* W1 = (const float*)d_in[5];
  const float* b1 = (const float*)d_in[6];
  const float* W2 = (const float*)d_in[7];
  const float* b2 = (const float*)d_in[8];
  float* out = (float*)d_out;

  const int B = out_size / 6;          // 32768
  const int T = in_sizes[0] / B;       // 1024
  const int blocks = B / 128;          // 8 waves/block * 16 rows/wave

  lstm_head_fused<<<blocks, 256, 0, stream>>>(x, W_ih, W_hh, b_ih, b_hh, W1,
                                              b1, W2, b2, out, T);
}